// PhaGatModel_48550310314397
// MI455X (gfx1250) — compile-verified
//
#include <hip/hip_runtime.h>
#include <hip/hip_bf16.h>

typedef __attribute__((ext_vector_type(2))) float v2f;
typedef __attribute__((ext_vector_type(8))) float v8f;

#define LRELU_SLOPE 0.2f
#define NEG_INF_F   (-1e9f)

// ---------------------------------------------------------------------------
// 1) Node embedding: hp[r] = (r==0) ? 0 : tf[r-1] @ W_emb + b_emb   [R=N+1,64]
// ---------------------------------------------------------------------------
__global__ void embed_kernel(const float* __restrict__ tf,
                             const float* __restrict__ W_emb,
                             const float* __restrict__ b_emb,
                             float* __restrict__ hp, int R) {
  int gid = blockIdx.x * blockDim.x + threadIdx.x;
  if (gid >= R * 64) return;
  int r = gid >> 6, c = gid & 63;
  float v = 0.f;
  if (r > 0) {
    const float* t = tf + (size_t)(r - 1) * 9;
    v = b_emb[c];
#pragma unroll
    for (int j = 0; j < 9; ++j) v += t[j] * W_emb[j * 64 + c];
  }
  hp[gid] = v;
}

// ---------------------------------------------------------------------------
// 2) Tiny fused-weight prep (single block, 64 threads):
//    Wcomb = W_dist @ Wm_gat            [10,64]
//    bcomb = b_dist @ Wm_gat            [64]
//    va2m  = Wm_gat @ a2                [64]   (a2 = a_gat[64:128])
//    wsm10 = W_dist @ va2m              [10]
//    bsm   = b_dist . va2m              scalar
// ---------------------------------------------------------------------------
__global__ void prep_small_kernel(const float* __restrict__ W_dist,
                                  const float* __restrict__ b_dist,
                                  const float* __restrict__ Wm_gat,
                                  const float* __restrict__ a_gat,
                                  float* __restrict__ Wcomb,
                                  float* __restrict__ bcomb,
                                  float* __restrict__ wsm10,
                                  float* __restrict__ bsm) {
  __shared__ float va2m[64];
  int t = threadIdx.x;  // 64 threads
  const float* a2 = a_gat + 64;
  float s = 0.f;
  for (int c2 = 0; c2 < 64; ++c2) s += Wm_gat[t * 64 + c2] * a2[c2];
  va2m[t] = s;
  __syncthreads();
  int c = t;
  float b = 0.f;
  for (int k = 0; k < 64; ++k) b += b_dist[k] * Wm_gat[k * 64 + c];
  bcomb[c] = b;
  for (int j = 0; j < 10; ++j) {
    float w = 0.f;
    for (int k = 0; k < 64; ++k) w += W_dist[j * 64 + k] * Wm_gat[k * 64 + c];
    Wcomb[j * 64 + c] = w;
  }
  if (t < 10) {
    float w = 0.f;
    for (int k = 0; k < 64; ++k) w += W_dist[t * 64 + k] * va2m[k];
    wsm10[t] = w;
  }
  if (t == 0) {
    float w = 0.f;
    for (int k = 0; k < 64; ++k) w += b_dist[k] * va2m[k];
    *bsm = w;
  }
}

// ---------------------------------------------------------------------------
// 3) Per-edge iter-0 attention scalar: sm0[e] = msg_in[e] . wsm10 + bsm
// ---------------------------------------------------------------------------
__global__ void edge_sm0_kernel(const float* __restrict__ fdg,
                                const float* __restrict__ rij,
                                const float* __restrict__ wsm10,
                                const float* __restrict__ bsm,
                                float* __restrict__ sm0, int E) {
  int e = blockIdx.x * blockDim.x + threadIdx.x;
  if (e >= E) return;
  float s = *bsm;
  const float* f = fdg + (size_t)e * 9;
#pragma unroll
  for (int j = 0; j < 9; ++j) s += f[j] * wsm10[j];
  s += rij[e] * wsm10[9];
  sm0[e] = s;
}

// ---------------------------------------------------------------------------
// 4) WMMA f32 GEMM:  C[R,64] = A[R,64] @ W[64,64]  via V_WMMA_F32_16X16X4_F32
//    One wave -> 16 rows x 64 cols (4 N-tiles of 16, K-loop of 16 x k=4).
//    A frag (16x4 f32, 2 VGPRs): lanes 0-15 K={0,1}, lanes 16-31 K={2,3}.
//    C frag (16x16 f32, 8 VGPRs): VGPR j -> row j (lanes 0-15) / row j+8.
//    Out-of-range A rows are clamped (not zeroed): a clamped lane only feeds
//    output rows >= R, which the store loop skips, so no predication needed.
// ---------------------------------------------------------------------------
__global__ void gemm64_wmma_kernel(const float* __restrict__ A,
                                   const float* __restrict__ W,
                                   float* __restrict__ C, int R) {
  int wave = threadIdx.x >> 5;
  int lane = threadIdx.x & 31;
  int half = lane >> 4;
  int lr   = lane & 15;
  int row0 = (blockIdx.x * 8 + wave) * 16;
  if (row0 >= R) return;  // wave-uniform exit; EXEC stays all-ones for WMMA

  v8f acc0 = {}, acc1 = {}, acc2 = {}, acc3 = {};
  int arow = row0 + lr;
  if (arow >= R) arow = R - 1;  // clamp: affected output rows are never stored
  const float* ap = A + (size_t)arow * 64 + (half ? 2 : 0);

  for (int kk = 0; kk < 16; ++kk) {
    int ka = kk * 4 + (half ? 2 : 0);
    v2f a = *(const v2f*)(ap + kk * 4);  // 8B-aligned unconditional load
    const float* w0 = W + ka * 64;
    const float* w1 = W + (ka + 1) * 64;
    v2f b0, b1, b2, b3;
    b0.x = w0[lr];      b0.y = w1[lr];
    b1.x = w0[16 + lr]; b1.y = w1[16 + lr];
    b2.x = w0[32 + lr]; b2.y = w1[32 + lr];
    b3.x = w0[48 + lr]; b3.y = w1[48 + lr];
    acc0 = __builtin_amdgcn_wmma_f32_16x16x4_f32(false, a, false, b0, (short)0, acc0, false, false);
    acc1 = __builtin_amdgcn_wmma_f32_16x16x4_f32(false, a, false, b1, (short)0, acc1, false, false);
    acc2 = __builtin_amdgcn_wmma_f32_16x16x4_f32(false, a, false, b2, (short)0, acc2, false, false);
    acc3 = __builtin_amdgcn_wmma_f32_16x16x4_f32(false, a, false, b3, (short)0, acc3, false, false);
  }

#pragma unroll
  for (int j = 0; j < 8; ++j) {
    int row = row0 + j + (half ? 8 : 0);
    if (row < R) {
      float* cp = C + (size_t)row * 64 + lr;
      cp[0]  = acc0[j];
      cp[16] = acc1[j];
      cp[32] = acc2[j];
      cp[48] = acc3[j];
    }
  }
}

// ---------------------------------------------------------------------------
// 5) Per-row dot with attention vector: out[r] = M[r,:] . avec
// ---------------------------------------------------------------------------
__global__ void rowdot_kernel(const float* __restrict__ M,
                              const float* __restrict__ avec,
                              float* __restrict__ outv, int R) {
  int r = blockIdx.x * blockDim.x + threadIdx.x;
  if (r >= R) return;
  const float4* mp = (const float4*)(M + (size_t)r * 64);
  const float4* av = (const float4*)avec;
  float s = 0.f;
#pragma unroll
  for (int i = 0; i < 16; ++i) {
    float4 m = mp[i], a = av[i];
    s += m.x * a.x + m.y * a.y + m.z * a.z + m.w * a.w;
  }
  outv[r] = s;
}

// ---------------------------------------------------------------------------
// 6) GAT aggregation: one wave32 per node.
//    ITER==0: Wm row computed on the fly from msg_in[e] @ Wcomb + bcomb
//    ITER==1: Wm row = mask(e) * Gm[scope_update[e]]
// ---------------------------------------------------------------------------
template <int ITER>
__global__ void gat_aggregate_kernel(const float* __restrict__ G,
                                     const float* __restrict__ sn,
                                     const float* __restrict__ sm0,
                                     const float* __restrict__ fdg,
                                     const float* __restrict__ rij,
                                     const float* __restrict__ Wcomb,
                                     const float* __restrict__ bcomb,
                                     const float* __restrict__ Gm,
                                     const float* __restrict__ smn,
                                     const int* __restrict__ scope_update,
                                     const int* __restrict__ slig,
                                     const int* __restrict__ see,
                                     const int* __restrict__ bsc,
                                     float* __restrict__ hp_out, int Nn) {
  __shared__ float sWcomb[640];
  __shared__ float sBcomb[64];
  __shared__ float sMsg[8][160];  // per-wave 16 slots x 10 feats

  int wave = threadIdx.x >> 5;
  int lane = threadIdx.x & 31;
  int n = blockIdx.x * 8 + wave;
  bool valid = n < Nn;
  int nn = valid ? n : 0;

  if (ITER == 0) {
    for (int i = threadIdx.x; i < 640; i += blockDim.x) sWcomb[i] = Wcomb[i];
    for (int i = threadIdx.x; i < 64; i += blockDim.x)  sBcomb[i] = bcomb[i];
  }

  int k = lane & 15;
  int in0 = see[(size_t)nn * 16 + k];
  int eb  = bsc[(size_t)nn * 16 + k];

  float sm_k;
  int   j1 = 0;
  float m1 = 0.f;
  if (ITER == 0) {
    sm_k = sm0[eb];
    if (lane < 16) {
      const float* f = fdg + (size_t)eb * 9;
#pragma unroll
      for (int j = 0; j < 9; ++j) sMsg[wave][k * 10 + j] = f[j];
      sMsg[wave][k * 10 + 9] = rij[eb];
    }
  } else {
    j1 = scope_update[eb];
    m1 = (slig[eb] > 0) ? 1.f : 0.f;
    sm_k = m1 * smn[j1];
  }
  if (ITER == 0) __syncthreads();  // uniform: no early returns above

  // logits + masked softmax over the 16 k-slots (replicated in both halves)
  float x = sn[in0] + sm_k;
  float logit = (x >= 0.f) ? x : LRELU_SLOPE * x;
  if (in0 == 0) logit = NEG_INF_F;
  float mx = logit;
#pragma unroll
  for (int d = 1; d < 16; d <<= 1) mx = fmaxf(mx, __shfl_xor(mx, d, 32));
  float ex = expf(logit - mx);
  float se = ex;
#pragma unroll
  for (int d = 1; d < 16; d <<= 1) se += __shfl_xor(se, d, 32);
  float alpha = ex / se;

  // each lane owns columns {2*lane, 2*lane+1}
  float accx = 0.f, accy = 0.f;
  for (int kk = 0; kk < 16; ++kk) {
    float a  = __shfl(alpha, kk, 32);
    int   ix = __shfl(in0, kk, 32);
    float2 wn = *(const float2*)(G + (size_t)ix * 64 + 2 * lane);
    float wmx, wmy;
    if (ITER == 0) {
      const float* mrow = &sMsg[wave][kk * 10];
      wmx = sBcomb[2 * lane];
      wmy = sBcomb[2 * lane + 1];
#pragma unroll
      for (int j = 0; j < 10; ++j) {
        float mv = mrow[j];
        wmx += mv * sWcomb[j * 64 + 2 * lane];
        wmy += mv * sWcomb[j * 64 + 2 * lane + 1];
      }
    } else {
      int   jj = __shfl(j1, kk, 32);
      float mm = __shfl(m1, kk, 32);
      float2 g = *(const float2*)(Gm + (size_t)jj * 64 + 2 * lane);
      wmx = mm * g.x;
      wmy = mm * g.y;
    }
    accx += a * (wn.x + wmx);
    accy += a * (wn.y + wmy);
  }

  if (valid) {
    float ox = accx > 0.f ? accx : expm1f(accx);  // elu, alpha=1
    float oy = accy > 0.f ? accy : expm1f(accy);
    hp_out[(size_t)(n + 1) * 64 + 2 * lane]     = ox;
    hp_out[(size_t)(n + 1) * 64 + 2 * lane + 1] = oy;
  }
  if (blockIdx.x == 0 && wave == 0) {  // keep pad row zero
    hp_out[2 * lane] = 0.f;
    hp_out[2 * lane + 1] = 0.f;
  }
}

// ---------------------------------------------------------------------------
// 7) Readout: out[b] = sum_m hp[l_scope[b,m]]   (one wave per molecule)
// ---------------------------------------------------------------------------
__global__ void readout_kernel(const float* __restrict__ hp,
                               const int* __restrict__ l_scope,
                               float* __restrict__ out, int MAXM) {
  int b = blockIdx.x;
  int lane = threadIdx.x;  // 32 threads
  float accx = 0.f, accy = 0.f;
  for (int m = 0; m < MAXM; ++m) {
    int idx = l_scope[(size_t)b * MAXM + m];
    float2 v = *(const float2*)(hp + (size_t)idx * 64 + 2 * lane);
    accx += v.x;
    accy += v.y;
  }
  out[(size_t)b * 64 + 2 * lane]     = accx;
  out[(size_t)b * 64 + 2 * lane + 1] = accy;
}

// ---------------------------------------------------------------------------
extern "C" void kernel_launch(void* const* d_in, const int* in_sizes, int n_in,
                              void* d_out, int out_size, void* d_ws, size_t ws_size,
                              hipStream_t stream) {
  const float* tf     = (const float*)d_in[0];
  const float* fdg    = (const float*)d_in[1];
  const float* rij    = (const float*)d_in[2];
  const int*   see    = (const int*)d_in[3];
  const int*   bsc    = (const int*)d_in[4];
  const int*   lsc    = (const int*)d_in[5];
  const int*   su     = (const int*)d_in[6];
  const int*   slig   = (const int*)d_in[7];
  const float* W_emb  = (const float*)d_in[8];
  const float* b_emb  = (const float*)d_in[9];
  const float* W_dist = (const float*)d_in[10];
  const float* b_dist = (const float*)d_in[11];
  const float* W_gat  = (const float*)d_in[12];
  const float* Wm_gat = (const float*)d_in[13];
  const float* a_gat  = (const float*)d_in[14];

  const int N    = in_sizes[0] / 9;   // 50000
  const int E    = in_sizes[2];       // 800000
  const int MAXM = 48;
  const int Bm   = in_sizes[5] / MAXM;
  const int R    = N + 1;

  float* ws = (float*)d_ws;
  size_t o = 0;
  float* hpA   = ws + o; o += (size_t)R * 64;
  float* hpB   = ws + o; o += (size_t)R * 64;
  float* G     = ws + o; o += (size_t)R * 64;
  float* Gm    = ws + o; o += (size_t)R * 64;
  float* sn    = ws + o; o += R;
  float* smn   = ws + o; o += R;
  float* sm0   = ws + o; o += E;
  float* Wcomb = ws + o; o += 640;
  float* bcomb = ws + o; o += 64;
  float* wsm10 = ws + o; o += 16;
  float* bsmp  = ws + o; o += 1;

  // Stage A: embeddings + fused small weights + per-edge scalar
  embed_kernel<<<(R * 64 + 255) / 256, 256, 0, stream>>>(tf, W_emb, b_emb, hpA, R);
  prep_small_kernel<<<1, 64, 0, stream>>>(W_dist, b_dist, Wm_gat, a_gat,
                                          Wcomb, bcomb, wsm10, bsmp);
  edge_sm0_kernel<<<(E + 255) / 256, 256, 0, stream>>>(fdg, rij, wsm10, bsmp, sm0, E);

  // Iteration 0: G = hp0 @ W_gat ; sn = G.a1 ; aggregate -> hp1 (hpB)
  gemm64_wmma_kernel<<<(R + 127) / 128, 256, 0, stream>>>(hpA, W_gat, G, R);
  rowdot_kernel<<<(R + 255) / 256, 256, 0, stream>>>(G, a_gat, sn, R);
  gat_aggregate_kernel<0><<<(N + 7) / 8, 256, 0, stream>>>(
      G, sn, sm0, fdg, rij, Wcomb, bcomb,
      nullptr, nullptr, nullptr, nullptr, see, bsc, hpB, N);

  // Iteration 1: G = hp1 @ W_gat ; Gm = hp1 @ Wm_gat ; aggregate -> hp2 (hpA)
  gemm64_wmma_kernel<<<(R + 127) / 128, 256, 0, stream>>>(hpB, W_gat, G, R);
  rowdot_kernel<<<(R + 255) / 256, 256, 0, stream>>>(G, a_gat, sn, R);
  gemm64_wmma_kernel<<<(R + 127) / 128, 256, 0, stream>>>(hpB, Wm_gat, Gm, R);
  rowdot_kernel<<<(R + 255) / 256, 256, 0, stream>>>(Gm, a_gat + 64, smn, R);
  gat_aggregate_kernel<1><<<(N + 7) / 8, 256, 0, stream>>>(
      G, sn, nullptr, nullptr, nullptr, nullptr, nullptr,
      Gm, smn, su, slig, see, bsc, hpA, N);

  // Readout
  readout_kernel<<<Bm, 32, 0, stream>>>(hpA, lsc, (float*)d_out, MAXM);
}